// LDIFLayer_48009144435097
// MI455X (gfx1250) — compile-verified
//
#include <hip/hip_runtime.h>
#include <math.h>

#define EC 32
#define NPTS 500000
#define NTILES (NPTS / 16)
#define DIV_EPS 1e-8f
#define THRESH 0.005f

#define BLOCKS 1024
#define TPB 256
#define NWAVES ((BLOCKS * TPB) / 32)

// workspace layout (floats)
#define WS_G_OFF 0      // 32 ellipsoids x 12 coeffs
#define WS_C_OFF 384    // 32 |constants|
#define WS_PEN_OFF 512  // NWAVES penalty partials

typedef float v2f __attribute__((ext_vector_type(2)));
typedef float v8f __attribute__((ext_vector_type(8)));

__device__ __forceinline__ float fast_sigmoid(float x) {
    return __builtin_amdgcn_rcpf(1.0f + __expf(-x));
}

// ---------------------------------------------------------------------------
// Prep: per-ellipsoid inv_cov -> quadratic-form coefficients (x -0.5 folded in)
// quad(x) = x^T M x - 2 x^T (M mu) + mu^T M mu,  M = R diag(1/(r+eps)) R^T
// -0.5*quad = g0*x^2+g1*y^2+g2*z^2 + g3*xy+g4*xz+g5*yz + g6*x+g7*y+g8*z + g9
// ---------------------------------------------------------------------------
__global__ void ldif_prep(const float* __restrict__ constants,
                          const float* __restrict__ centers,
                          const float* __restrict__ radii,
                          const float* __restrict__ rots,
                          float* __restrict__ ws) {
    int e = threadIdx.x;
    if (e >= EC) return;

    float c = fabsf(constants[e]);
    float r = fabsf(radii[e]);
    float dinv = 1.0f / (r + DIV_EPS);

    float rx = rots[e * 3 + 0], ry = rots[e * 3 + 1], rz = rots[e * 3 + 2];
    float cx = cosf(rx), sx = sinf(rx);
    float cy = cosf(ry), sy = sinf(ry);
    float cz = cosf(rz), sz = sinf(rz);

    float R[3][3];
    R[0][0] = cz * cy; R[0][1] = cz * sy * sx - sz * cx; R[0][2] = cz * sy * cx + sz * sx;
    R[1][0] = sz * cy; R[1][1] = sz * sy * sx + cz * cx; R[1][2] = sz * sy * cx - cz * sx;
    R[2][0] = -sy;     R[2][1] = cy * sx;                R[2][2] = cy * cx;

    float M[3][3];
#pragma unroll
    for (int i = 0; i < 3; ++i)
#pragma unroll
        for (int k = 0; k < 3; ++k)
            M[i][k] = dinv * (R[i][0] * R[k][0] + R[i][1] * R[k][1] + R[i][2] * R[k][2]);

    float mu0 = centers[e * 3 + 0], mu1 = centers[e * 3 + 1], mu2 = centers[e * 3 + 2];
    float Mmu0 = M[0][0] * mu0 + M[0][1] * mu1 + M[0][2] * mu2;
    float Mmu1 = M[1][0] * mu0 + M[1][1] * mu1 + M[1][2] * mu2;
    float Mmu2 = M[2][0] * mu0 + M[2][1] * mu1 + M[2][2] * mu2;
    float Q = mu0 * Mmu0 + mu1 * Mmu1 + mu2 * Mmu2;

    float* g = ws + WS_G_OFF + e * 12;
    g[0] = -0.5f * M[0][0];
    g[1] = -0.5f * M[1][1];
    g[2] = -0.5f * M[2][2];
    g[3] = -M[0][1];
    g[4] = -M[0][2];
    g[5] = -M[1][2];
    g[6] = Mmu0;
    g[7] = Mmu1;
    g[8] = Mmu2;
    g[9] = -0.5f * Q;
    g[10] = 0.0f;
    g[11] = 0.0f;
    ws[WS_C_OFF + e] = c;
}

// ---------------------------------------------------------------------------
// Main: one wave per 16-point tile. WMMA f32 16x16x4 computes -0.5*quad for
// 32 ellipsoids x 16 points (2 M-tiles x 3 K-chunks). Streams activations
// exactly once with coalesced float4 loads. No atomics (deterministic).
//
// B-operand per 16x16x4: lanes 0-15 supply K={0,1} in (x,y), lanes 16-31
// supply K={2,3}. Feature order: [x2,y2,z2,xy | xz,yz,x,y | z,1,0,0].
// ---------------------------------------------------------------------------
__global__ __launch_bounds__(TPB) void ldif_main(const float* __restrict__ wsp,
                                                 const float* __restrict__ act,
                                                 const float* __restrict__ dists,
                                                 const float* __restrict__ ws,
                                                 float* __restrict__ out,
                                                 float* __restrict__ pen_partials) {
    const int lane = threadIdx.x & 31;
    const int wid = (blockIdx.x * blockDim.x + threadIdx.x) >> 5;
    const int col = lane & 15;   // point within tile (N index of WMMA)
    const int half = lane >> 4;  // lane half selects K pair / ellipsoid subset
    const bool hi = (half != 0);
    const int koff = 2 * half;

    // Loop-invariant A-matrix operands: row M = col, K split by lane half.
    v2f A0[3], A1[3];
    {
        const float* g0 = ws + WS_G_OFF + (size_t)col * 12;
        const float* g1 = ws + WS_G_OFF + (size_t)(col + 16) * 12;
#pragma unroll
        for (int ch = 0; ch < 3; ++ch) {
            A0[ch].x = g0[4 * ch + koff];
            A0[ch].y = g0[4 * ch + koff + 1];
            A1[ch].x = g1[4 * ch + koff];
            A1[ch].y = g1[4 * ch + koff + 1];
        }
    }
    // |c| for the 16 ellipsoids this lane owns in the D layout:
    // acc0[v] -> e = 8*half + v ; acc1[v] -> e = 16 + 8*half + v
    float cA[8], cB[8];
#pragma unroll
    for (int v = 0; v < 8; ++v) {
        cA[v] = ws[WS_C_OFF + 8 * half + v];
        cB[v] = ws[WS_C_OFF + 16 + 8 * half + v];
    }

    const int nwaves = (gridDim.x * blockDim.x) >> 5;
    float pen_acc = 0.0f;

    for (int tile = wid; tile < NTILES; tile += nwaves) {
        const int n = tile * 16 + col;
        const float x = wsp[3 * (size_t)n + 0];
        const float y = wsp[3 * (size_t)n + 1];
        const float z = wsp[3 * (size_t)n + 2];

        // B operands: one cndmask per element instead of dynamic array index.
        v2f b0, b1, b2;
        b0.x = hi ? z * z : x * x;
        b0.y = hi ? x * y : y * y;
        b1.x = hi ? x : x * z;
        b1.y = hi ? y : y * z;
        b2.x = hi ? 0.0f : z;
        b2.y = hi ? 0.0f : 1.0f;

        v8f acc0 = {0.f, 0.f, 0.f, 0.f, 0.f, 0.f, 0.f, 0.f};
        v8f acc1 = {0.f, 0.f, 0.f, 0.f, 0.f, 0.f, 0.f, 0.f};
        acc0 = __builtin_amdgcn_wmma_f32_16x16x4_f32(false, A0[0], false, b0,
                                                     (short)0, acc0, false, false);
        acc1 = __builtin_amdgcn_wmma_f32_16x16x4_f32(false, A1[0], false, b0,
                                                     (short)0, acc1, false, false);
        acc0 = __builtin_amdgcn_wmma_f32_16x16x4_f32(false, A0[1], false, b1,
                                                     (short)0, acc0, false, false);
        acc1 = __builtin_amdgcn_wmma_f32_16x16x4_f32(false, A1[1], false, b1,
                                                     (short)0, acc1, false, false);
        acc0 = __builtin_amdgcn_wmma_f32_16x16x4_f32(false, A0[2], false, b2,
                                                     (short)0, acc0, false, false);
        acc1 = __builtin_amdgcn_wmma_f32_16x16x4_f32(false, A1[2], false, b2,
                                                     (short)0, acc1, false, false);

        // rbf = |c| * exp(-0.5*quad); partial sums over this lane's 16 ellipsoids
        float rbf0[8], rbf1[8];
        float s = 0.0f, pen = 0.0f;
#pragma unroll
        for (int v = 0; v < 8; ++v) {
            rbf0[v] = cA[v] * __expf(acc0[v]);
            rbf1[v] = cB[v] * __expf(acc1[v]);
            s += rbf0[v] + rbf1[v];
            pen += fabsf(rbf0[v] - THRESH) + fabsf(rbf1[v] - THRESH);
        }
        pen_acc += pen;  // each (e,n) counted exactly once across the wave

        // partner lane (lane ^ 16) holds the other 16 ellipsoids of point n
        s += __shfl_xor(s, 16, 32);
        const float inv = __builtin_amdgcn_rcpf(s + 0.01f);
        const float dist = dists[n];

        float o0 = 0.f, o1 = 0.f, o2 = 0.f, o3 = 0.f;
#pragma unroll
        for (int v = 0; v < 8; ++v) {
            const int e0 = 8 * half + v;
            const float4 a4 = *(const float4*)(act + 4 * ((size_t)e0 * NPTS + n));
            float w = rbf0[v] * inv;
            o0 += w * fast_sigmoid(a4.x);
            o1 += w * fast_sigmoid(a4.y);
            o2 += w * fast_sigmoid(a4.z);
            o3 += w * (1.0f - __expf(-fmaxf(a4.w, 0.0f) * dist));

            const int e1 = e0 + 16;
            const float4 b4 = *(const float4*)(act + 4 * ((size_t)e1 * NPTS + n));
            w = rbf1[v] * inv;
            o0 += w * fast_sigmoid(b4.x);
            o1 += w * fast_sigmoid(b4.y);
            o2 += w * fast_sigmoid(b4.z);
            o3 += w * (1.0f - __expf(-fmaxf(b4.w, 0.0f) * dist));
        }
        o0 += __shfl_xor(o0, 16, 32);
        o1 += __shfl_xor(o1, 16, 32);
        o2 += __shfl_xor(o2, 16, 32);
        o3 += __shfl_xor(o3, 16, 32);

        if (half == 0) {
            float4 r4;
            r4.x = o0; r4.y = o1; r4.z = o2; r4.w = o3;
            *(float4*)(out + 4 * (size_t)n) = r4;
        }
    }

    // deterministic per-wave penalty partial (every wave writes, even if idle)
#pragma unroll
    for (int off = 16; off > 0; off >>= 1) pen_acc += __shfl_xor(pen_acc, off, 32);
    if (lane == 0) pen_partials[wid] = pen_acc;
}

// ---------------------------------------------------------------------------
// Fixed-order reduction of per-wave penalty partials -> mean
// ---------------------------------------------------------------------------
__global__ __launch_bounds__(256) void ldif_pen_reduce(const float* __restrict__ pen_partials,
                                                       float* __restrict__ pen_out,
                                                       int nwaves) {
    __shared__ float sm[256];
    float s = 0.0f;
    for (int i = threadIdx.x; i < nwaves; i += 256) s += pen_partials[i];
    sm[threadIdx.x] = s;
    __syncthreads();
#pragma unroll
    for (int off = 128; off > 0; off >>= 1) {
        if ((int)threadIdx.x < off) sm[threadIdx.x] += sm[threadIdx.x + off];
        __syncthreads();
    }
    if (threadIdx.x == 0) pen_out[0] = sm[0] * (1.0f / (float)NPTS);
}

extern "C" void kernel_launch(void* const* d_in, const int* in_sizes, int n_in,
                              void* d_out, int out_size, void* d_ws, size_t ws_size,
                              hipStream_t stream) {
    const float* wsp = (const float*)d_in[0];       // (N,3)
    const float* act = (const float*)d_in[1];       // (EC,N,4)
    const float* dists = (const float*)d_in[2];     // (N,1)
    const float* constants = (const float*)d_in[3]; // (EC,1)
    const float* centers = (const float*)d_in[4];   // (EC,3)
    const float* radii = (const float*)d_in[5];     // (EC,1)
    const float* rots = (const float*)d_in[6];      // (EC,3)

    float* out = (float*)d_out;  // N*4 outputs, then penalty scalar
    float* ws = (float*)d_ws;

    ldif_prep<<<1, 32, 0, stream>>>(constants, centers, radii, rots, ws);
    ldif_main<<<BLOCKS, TPB, 0, stream>>>(wsp, act, dists, ws, out, ws + WS_PEN_OFF);
    ldif_pen_reduce<<<1, 256, 0, stream>>>(ws + WS_PEN_OFF, out + (out_size - 1), NWAVES);
}